// FeatureExpansion_28810640621987
// MI455X (gfx1250) — compile-verified
//
#include <hip/hip_runtime.h>

typedef __attribute__((ext_vector_type(2))) float v2f;
typedef __attribute__((ext_vector_type(8))) float v8f;

#define STRIDE 10
#define NF 32
#define N_OL 60
#define OUT_PER_TILE 1120   // 4*32 + 2*496
#define NPAIR 496

__device__ __forceinline__ float div_no_nan(float a, float b) {
    return (b == 0.0f) ? 0.0f : a / b;
}

__global__ __launch_bounds__(256) void feat_expand_kernel(
    const float* __restrict__ in, float* __restrict__ out, int nTiles)
{
    const int lane = threadIdx.x & 31;
    const int wave = threadIdx.x >> 5;
    const int tile = blockIdx.x * 8 + wave;
    if (tile >= nTiles) return;          // wave-uniform: EXEC stays all-ones

    const int lo = lane & 15;
    const int hi = lane >> 4;            // 0 or 1

    const float* tp = in + (size_t)tile * (STRIDE * NF);

    // Lane owns feature f = lane. 10 loads, each 32 lanes x 4B = one 128B line.
    float xs[STRIDE];
#pragma unroll
    for (int s = 0; s < STRIDE; ++s)
        xs[s] = tp[s * NF + lane];

    float sum = 0.f, ldsum = 0.f;
#pragma unroll
    for (int s = 0; s < STRIDE; ++s) {
        sum   += xs[s];
        ldsum += xs[s] * (float)(s + 1);
    }
    const float mean = sum * 0.1f;
    float ms[STRIDE];
    float vsum = 0.f;
#pragma unroll
    for (int s = 0; s < STRIDE; ++s) {
        ms[s] = xs[s] - mean;
        vsum += ms[s] * ms[s];
    }
    const float stdv = sqrtf(vsum * 0.1f);
    const float z    = div_no_nan(mean, stdv);
    const float ld   = ldsum * (1.0f / 55.0f);          // linspace(1,10)/55
    const float ret  = div_no_nan(xs[9], xs[0]) - 1.0f;
    // Guarded reciprocal: keeps divide_no_nan zero semantics for corr while
    // turning 496 per-tile divides into 2 multiplies per element.
    const float rstd = (stdv == 0.0f) ? 0.0f : (1.0f / stdv);

    float* op = out + (size_t)tile * OUT_PER_TILE;
    op[lane]      = stdv;
    op[32 + lane] = z;
    op[64 + lane] = ld;
    op[96 + lane] = ret;

    // ---- Fragments for G = ms^T (32x10) * ms (10x32), K padded to 12 ----
    // V_WMMA_F32_16X16X4_F32 A-layout: lane<16 holds {A[M][K0],A[M][K0+1]},
    // lane>=16 holds {A[M][K0+2],A[M][K0+3]}.  For the symmetric Gram the
    // B fragment is identical, so build one fragment per (half, K-chunk).
    v2f frag[2][3];
#pragma unroll
    for (int h = 0; h < 2; ++h) {
        const int src = h * 16 + lo;     // feature whose column we need
#pragma unroll
        for (int c = 0; c < 2; ++c) {
            float p0 = __shfl(ms[c * 4 + 0], src, 32);
            float p1 = __shfl(ms[c * 4 + 1], src, 32);
            float p2 = __shfl(ms[c * 4 + 2], src, 32);
            float p3 = __shfl(ms[c * 4 + 3], src, 32);
            frag[h][c].x = hi ? p2 : p0;
            frag[h][c].y = hi ? p3 : p1;
        }
        {   // chunk 2: K = 8,9 real; 10,11 are zero padding
            float p0 = __shfl(ms[8], src, 32);
            float p1 = __shfl(ms[9], src, 32);
            frag[h][2].x = hi ? 0.f : p0;
            frag[h][2].y = hi ? 0.f : p1;
        }
    }

    v8f d00 = {}, d10 = {}, d11 = {};
#pragma unroll
    for (int c = 0; c < 3; ++c) {
        d00 = __builtin_amdgcn_wmma_f32_16x16x4_f32(false, frag[0][c], false, frag[0][c], (short)0, d00, false, false);
        d10 = __builtin_amdgcn_wmma_f32_16x16x4_f32(false, frag[1][c], false, frag[0][c], (short)0, d10, false, false);
        d11 = __builtin_amdgcn_wmma_f32_16x16x4_f32(false, frag[1][c], false, frag[1][c], (short)0, d11, false, false);
    }

    // reciprocal stds of the rows/cols this lane's D elements touch
    float rr0[8], rr1[8];
#pragma unroll
    for (int v = 0; v < 8; ++v) {
        rr0[v] = __shfl(rstd, v + hi * 8, 32);
        rr1[v] = __shfl(rstd, 16 + v + hi * 8, 32);
    }
    const float rc0 = __shfl(rstd, lo, 32);
    const float rc1 = __shfl(rstd, 16 + lo, 32);

    float* covp  = op + 128;
    float* corrp = op + 128 + NPAIR;

    // D element in VGPR v of tile (ti,tj): r = ti*16 + v + hi*8, c = tj*16 + lo
#pragma unroll
    for (int v = 0; v < 8; ++v) {
        const int M = v + hi * 8;
        {   // tile (0,0): strict lower triangle
            int r = M, c = lo;
            if (r > c) {
                int idx = (r * (r - 1)) / 2 + c;
                float g = d00[v];
                covp[idx]  = g * (1.0f / 9.0f);
                corrp[idx] = (g * 0.1f) * (rr0[v] * rc0);
            }
        }
        {   // tile (1,0): fully below diagonal
            int r = 16 + M, c = lo;
            int idx = (r * (r - 1)) / 2 + c;
            float g = d10[v];
            covp[idx]  = g * (1.0f / 9.0f);
            corrp[idx] = (g * 0.1f) * (rr1[v] * rc0);
        }
        {   // tile (1,1): strict lower triangle
            int r = 16 + M, c = 16 + lo;
            if (r > c) {
                int idx = (r * (r - 1)) / 2 + c;
                float g = d11[v];
                covp[idx]  = g * (1.0f / 9.0f);
                corrp[idx] = (g * 0.1f) * (rr1[v] * rc1);
            }
        }
    }
}

extern "C" void kernel_launch(void* const* d_in, const int* in_sizes, int n_in,
                              void* d_out, int out_size, void* d_ws, size_t ws_size,
                              hipStream_t stream) {
    (void)n_in; (void)out_size; (void)d_ws; (void)ws_size;
    const float* in = (const float*)d_in[0];
    float* out = (float*)d_out;
    const int n = in_sizes[0];                 // B * 600 * 32
    const int B = n / (600 * 32);
    const int nTiles = B * N_OL;               // one wave per (b, ol)
    const int blocks = (nTiles + 7) / 8;       // 8 waves (256 threads) per block
    feat_expand_kernel<<<blocks, 256, 0, stream>>>(in, out, nTiles);
}